// ProtoDot_4578435138156
// MI455X (gfx1250) — compile-verified
//
#include <hip/hip_runtime.h>
#include <hip/hip_bf16.h>

// Problem dims (fixed by setup_inputs):
// B=8, N=5, K=5, Q=4, L=128, F=768, NQ=N*Q=20, T=2N+1=11 (pad to 16)
#define BB     8
#define NN     5
#define KK     5
#define LL     128
#define FF     768
#define NQ     20
#define TT     11
#define TPAD   16
#define MROWS  (NQ * LL)          // 2560 rows per batch
#define MTILES (MROWS / 16)       // 160 tiles per batch
#define KSTEPS (FF / 4)           // 192 WMMA k-steps
#define NROWS  (BB * MROWS)       // 20480 total rows
#define LOSS_BLOCKS (NROWS / 256) // 80

typedef float v2f __attribute__((ext_vector_type(2)));
typedef float v8f __attribute__((ext_vector_type(8)));

// ---------------------------------------------------------------------------
// Kernel 1: prototypes. grid = (B, F/256), block = 256. Each thread owns one f.
// proto_pad layout: (B, 16, F) row-major, rows 11..15 zero.
// ---------------------------------------------------------------------------
__global__ __launch_bounds__(256) void proto_kernel(
    const float* __restrict__ sup, const int* __restrict__ Bm,
    const int* __restrict__ Im, float* __restrict__ proto) {
  const int b = blockIdx.x;
  const int f = blockIdx.y * 256 + threadIdx.x;

  float accB[NN], accI[NN], cntB[NN], cntI[NN];
  float accO = 0.f, cntO = 0.f;
#pragma unroll
  for (int n = 0; n < NN; ++n) { accB[n] = accI[n] = cntB[n] = cntI[n] = 0.f; }

  for (int n = 0; n < NN; ++n) {
    for (int k = 0; k < KK; ++k) {
      const int   mbase = ((b * NN + n) * KK + k) * LL;
      const float* sp   = sup + (size_t)mbase * FF + f;
      const int*   bmp  = Bm + mbase;
      const int*   imp  = Im + mbase;
      for (int l = 0; l < LL; ++l) {
        const float v  = sp[(size_t)l * FF];
        const float bm = (float)bmp[l];   // uniform across block -> scalar load
        const float im = (float)imp[l];
        const float om = 1.f - bm - im;
        accB[n] += v * bm;  accI[n] += v * im;  accO += v * om;
        cntB[n] += bm;      cntI[n] += im;      cntO += om;
      }
    }
  }

  const float EPS = 1e-8f;
  float* pb = proto + (size_t)b * TPAD * FF;
  pb[f] = accO / (cntO + EPS);                      // tag 0 = O
#pragma unroll
  for (int n = 0; n < NN; ++n) {
    pb[(size_t)(2 * n + 1) * FF + f] = accB[n] / (cntB[n] + EPS);
    pb[(size_t)(2 * n + 2) * FF + f] = accI[n] / (cntI[n] + EPS);
  }
#pragma unroll
  for (int t = TT; t < TPAD; ++t) pb[(size_t)t * FF + f] = 0.f;
}

// ---------------------------------------------------------------------------
// Kernel 2: logits GEMM via V_WMMA_F32_16X16X4_F32.
// grid = B*MTILES one-wave blocks. D(16x16) = Q_tile(16x768) x Proto^T(768x16).
// A 16x4 layout: lanes 0-15 -> (K=0,1), lanes 16-31 -> (K=2,3), M = lane%16.
// B 4x16 layout: mirrored, N = lane%16.
// D: VGPR r, lane l -> (M = r + 8*(l>=16), N = l&15).
// ---------------------------------------------------------------------------
__global__ __launch_bounds__(32) void logits_wmma_kernel(
    const float* __restrict__ qry, const float* __restrict__ proto,
    float* __restrict__ logits) {
  const int tile = blockIdx.x;
  const int b    = tile / MTILES;
  const int mt   = tile % MTILES;
  const int lane = threadIdx.x;
  const bool hi  = lane >= 16;
  const int  lr  = hi ? lane - 16 : lane;
  const int  koff = hi ? 2 : 0;

  const float* qrow = qry + (size_t)b * MROWS * FF + (size_t)(mt * 16 + lr) * FF + koff;
  const float* prow = proto + (size_t)b * TPAD * FF + (size_t)lr * FF + koff;

  v8f acc = {};
  for (int k4 = 0; k4 < KSTEPS; ++k4) {
    v2f a = *(const v2f*)(qrow + k4 * 4);
    v2f p = *(const v2f*)(prow + k4 * 4);
    acc = __builtin_amdgcn_wmma_f32_16x16x4_f32(
        /*neg_a=*/false, a, /*neg_b=*/false, p,
        /*c_mod=*/(short)0, acc, /*reuse_a=*/false, /*reuse_b=*/false);
  }

  const int col = lane & 15;
  if (col < TT) {
    float* out = logits + (size_t)(b * MROWS + mt * 16) * TT + col;
    const int mo = hi ? 8 : 0;
#pragma unroll
    for (int r = 0; r < 8; ++r) out[(size_t)(r + mo) * TT] = acc[r];
  }
}

// ---------------------------------------------------------------------------
// Kernel 3: per-row argmax + log-softmax loss. grid = 80 x 256 threads.
// Deterministic: block partials to workspace, no float atomics.
// ---------------------------------------------------------------------------
__global__ __launch_bounds__(256) void loss_pred_kernel(
    const float* __restrict__ logits, const int* __restrict__ label,
    float* __restrict__ pred_out, float* __restrict__ partial) {
  __shared__ float sm[256];
  const int r = blockIdx.x * 256 + threadIdx.x;   // < NROWS

  const float* row = logits + (size_t)r * TT;
  float mx = row[0];
  int arg = 0;
#pragma unroll
  for (int t = 1; t < TT; ++t) {
    const float v = row[t];
    if (v > mx) { mx = v; arg = t; }               // first-max tie rule
  }
  float se = 0.f;
#pragma unroll
  for (int t = 0; t < TT; ++t) se += expf(row[t] - mx);
  const int lab = label[r];
  const float li = -(row[lab] - mx - logf(se));
  pred_out[r] = (float)arg;

  sm[threadIdx.x] = li;
  __syncthreads();
  for (int s = 128; s > 0; s >>= 1) {
    if (threadIdx.x < s) sm[threadIdx.x] += sm[threadIdx.x + s];
    __syncthreads();
  }
  if (threadIdx.x == 0) partial[blockIdx.x] = sm[0];
}

__global__ void finalize_kernel(const float* __restrict__ partial,
                                float* __restrict__ out) {
  if (threadIdx.x == 0) {
    float s = 0.f;
    for (int i = 0; i < LOSS_BLOCKS; ++i) s += partial[i];
    out[0] = s / (float)NROWS;
  }
}

// ---------------------------------------------------------------------------
// d_out layout (fp32): [ loss(1) | logits(B*NQ*L*T = 225280) | pred(20480) ]
// d_ws  layout (fp32): [ proto_pad(B*16*F = 98304) | partials(80) ]
// ---------------------------------------------------------------------------
extern "C" void kernel_launch(void* const* d_in, const int* in_sizes, int n_in,
                              void* d_out, int out_size, void* d_ws, size_t ws_size,
                              hipStream_t stream) {
  (void)in_sizes; (void)n_in; (void)out_size; (void)ws_size;
  const float* sup = (const float*)d_in[0];
  const float* qry = (const float*)d_in[1];
  const int*   Bm  = (const int*)d_in[2];
  const int*   Im  = (const int*)d_in[3];
  const int*   lab = (const int*)d_in[4];

  float* out      = (float*)d_out;
  float* loss_out = out;
  float* logits   = out + 1;
  float* pred     = out + 1 + (size_t)BB * MROWS * TT;

  float* proto    = (float*)d_ws;
  float* partial  = proto + (size_t)BB * TPAD * FF;

  proto_kernel<<<dim3(BB, FF / 256), 256, 0, stream>>>(sup, Bm, Im, proto);
  logits_wmma_kernel<<<BB * MTILES, 32, 0, stream>>>(qry, proto, logits);
  loss_pred_kernel<<<LOSS_BLOCKS, 256, 0, stream>>>(logits, lab, pred, partial);
  finalize_kernel<<<1, 32, 0, stream>>>(partial, loss_out);
}